// YOLOXHeadScript_75325136437522
// MI455X (gfx1250) — compile-verified
//
#include <hip/hip_runtime.h>
#include <hip/hip_bf16.h>
#include <cmath>

typedef __attribute__((ext_vector_type(2))) float v2f;
typedef __attribute__((ext_vector_type(8))) float v8f;

#define KK 10

__device__ __forceinline__ float sigm(float x) { return 1.0f / (1.0f + expf(-x)); }

__device__ __forceinline__ float pair_iou(float gx, float gy, float gw, float gh,
                                          float px, float py, float pw, float ph) {
    float gtlx = gx - 0.5f * gw, gtly = gy - 0.5f * gh;
    float gbrx = gx + 0.5f * gw, gbry = gy + 0.5f * gh;
    float ptlx = px - 0.5f * pw, ptly = py - 0.5f * ph;
    float pbrx = px + 0.5f * pw, pbry = py + 0.5f * ph;
    float tlx = fmaxf(gtlx, ptlx), tly = fmaxf(gtly, ptly);
    float brx = fminf(gbrx, pbrx), bry = fminf(gbry, pbry);
    float wx = fmaxf(brx - tlx, 0.0f), wy = fmaxf(bry - tly, 0.0f);
    float inter = wx * wy;
    float uni = gw * gh + pw * ph - inter;
    return inter / uni;
}

__device__ __forceinline__ void box_ctr(float gx, float gy, float gw, float gh,
                                        float cx, float cy, float r,
                                        bool& inb, bool& inc) {
    float m1 = fminf(fminf(cx - (gx - 0.5f * gw), (gx + 0.5f * gw) - cx),
                     fminf(cy - (gy - 0.5f * gh), (gy + 0.5f * gh) - cy));
    inb = m1 > 0.0f;
    float m2 = fminf(fminf(cx - (gx - r), (gx + r) - cx),
                     fminf(cy - (gy - r), (gy + r) - cy));
    inc = m2 > 0.0f;
}

__device__ __forceinline__ float pair_cost(float gx, float gy, float gw, float gh,
                                           float px, float py, float pw, float ph,
                                           float cx, float cy, float stride,
                                           float negsum, float sobj, float cls_logit,
                                           int fga, float* iou_out) {
    float iou = pair_iou(gx, gy, gw, gh, px, py, pw, ph);
    bool inb, inc;
    box_ctr(gx, gy, gw, gh, cx, cy, 2.5f * stride, inb, inc);
    bool geom = inb && inc;
    float pos = sqrtf(sigm(cls_logit) * sobj);
    float pt = fmaxf(logf(1.0f - pos), -100.0f) - fmaxf(logf(pos), -100.0f);
    float c = negsum + pt + 3.0f * (-logf(iou + 1e-8f)) + (geom ? 0.0f : 100000.0f);
    if (!fga) c = 1e15f;
    *iou_out = iou;
    return c;
}

// ---------------------------------------------------------------------------
// Kernel A: per (b,a) geometric fg mask, obj sigmoid, init match scratch.
// ---------------------------------------------------------------------------
__global__ void k_precompute(const float* __restrict__ labels,
                             const float* __restrict__ objp,
                             const float* __restrict__ strides,
                             const float* __restrict__ xs,
                             const float* __restrict__ ys,
                             int* __restrict__ fg, float* __restrict__ sobj,
                             int* __restrict__ mcnt, int* __restrict__ ming,
                             int Bn, int Gn, int An) {
    int t = blockIdx.x * blockDim.x + threadIdx.x;
    if (t >= Bn * An) return;
    int b = t / An, a = t - b * An;
    float s = strides[a];
    float cx = (xs[a] + 0.5f) * s, cy = (ys[a] + 0.5f) * s;
    float r = 2.5f * s;
    bool anyb = false, anyc = false;
    const float* L = labels + (size_t)b * Gn * 5;
    for (int g = 0; g < Gn; ++g) {
        float gx = L[g * 5 + 1], gy = L[g * 5 + 2];
        float gw = L[g * 5 + 3], gh = L[g * 5 + 4];
        bool ib, ic;
        box_ctr(gx, gy, gw, gh, cx, cy, r, ib, ic);
        anyb |= ib;
        anyc |= ic;
    }
    fg[t] = (anyb || anyc) ? 1 : 0;
    sobj[t] = sigm(objp[t]);
    mcnt[t] = 0;
    ming[t] = 0x7fffffff;
}

// ---------------------------------------------------------------------------
// Kernel B: neg_sum[a] = sum_c -max(log(1 - sqrt(sig_c*sig_obj)), -100)
// Reduction over NC classes with V_WMMA_F32_16X16X4_F32 (B = all-ones, f32
// accumulate -> every D column is the exact row sum; class->K-slot mapping is
// arbitrary since the sum is slot-order independent).
// Memory path: one aligned b128 load per lane feeds TWO WMMA chunks:
//   lanes 0-15  load classes [8j .. 8j+3]  (K0,K1 of chunk0; K0,K1 of chunk1)
//   lanes 16-31 load classes [8j+4..8j+7]  (K2,K3 of chunk0; K2,K3 of chunk1)
// One wave per tile of 16 anchors; full-EXEC around the WMMAs.
// ---------------------------------------------------------------------------
__global__ void k_negsum(const float* __restrict__ clsp,
                         const float* __restrict__ sobj,
                         float* __restrict__ negsum,
                         int Bn, int An, int NC) {
    int tiles = (An + 15) / 16;
    int blk = blockIdx.x;
    int b = blk / tiles, tile = blk - b * tiles;
    int lane = threadIdx.x;
    int m = lane & 15;
    int half = lane >> 4;
    int anchor = tile * 16 + m;
    if (anchor >= An) anchor = An - 1;  // clamp loads; stores guarded below
    size_t base = ((size_t)(b * An + anchor)) * NC;
    float so = sobj[b * An + anchor];

    v8f acc = {};
    int kc = 0;
    for (; kc + 7 < NC; kc += 8) {
        const float4 q = *(const float4*)(clsp + base + kc + half * 4);
        float s0 = sqrtf(sigm(q.x) * so);
        float s1 = sqrtf(sigm(q.y) * so);
        float s2 = sqrtf(sigm(q.z) * so);
        float s3 = sqrtf(sigm(q.w) * so);
        float n0 = -fmaxf(logf(1.0f - s0), -100.0f);
        float n1 = -fmaxf(logf(1.0f - s1), -100.0f);
        float n2 = -fmaxf(logf(1.0f - s2), -100.0f);
        float n3 = -fmaxf(logf(1.0f - s3), -100.0f);
        v2f bv; bv.x = 1.0f; bv.y = 1.0f;
        v2f av; av.x = n0; av.y = n1;
        acc = __builtin_amdgcn_wmma_f32_16x16x4_f32(false, av, false, bv,
                                                    (short)0, acc, false, false);
        av.x = n2; av.y = n3;
        acc = __builtin_amdgcn_wmma_f32_16x16x4_f32(false, av, false, bv,
                                                    (short)0, acc, false, false);
    }
    for (; kc + 3 < NC; kc += 4) {  // scalar tail (NC % 8 != 0)
        int c0 = kc + half * 2;
        float x0 = clsp[base + c0];
        float x1 = clsp[base + c0 + 1];
        float s0 = sqrtf(sigm(x0) * so);
        float s1 = sqrtf(sigm(x1) * so);
        float n0 = -fmaxf(logf(1.0f - s0), -100.0f);
        float n1 = -fmaxf(logf(1.0f - s1), -100.0f);
        v2f av; av.x = n0; av.y = n1;
        v2f bv; bv.x = 1.0f; bv.y = 1.0f;
        acc = __builtin_amdgcn_wmma_f32_16x16x4_f32(false, av, false, bv,
                                                    (short)0, acc, false, false);
    }
    // D layout (16x16 f32): VGPR r: lanes 0-15 -> M=r, lanes 16-31 -> M=8+r.
    int sel = -1, tgt = -1;
    if (lane < 8) { sel = lane; tgt = tile * 16 + lane; }
    else if (lane >= 24) { sel = lane - 24; tgt = tile * 16 + 8 + (lane - 24); }
    if (sel >= 0 && tgt < An) {
        float v = acc[0];
#pragma unroll
        for (int i = 1; i < 8; ++i)
            if (sel == i) v = acc[i];
        negsum[b * An + tgt] = v;
    }
}

// ---------------------------------------------------------------------------
// Kernel C: one wave per (b,g). Stream over anchors computing cost & iou,
// keep per-lane top-10 (min cost, lexicographic on (cost,idx) to match JAX
// stable top_k) and top-10 ious_fg; merge via LDS; dyn_k = clip(trunc(sum),1,K);
// scatter matches with atomics.
// ---------------------------------------------------------------------------
__global__ void k_assign(const float* __restrict__ labels,
                         const float* __restrict__ bbox,
                         const float* __restrict__ clsp,
                         const float* __restrict__ strides,
                         const float* __restrict__ xs,
                         const float* __restrict__ ys,
                         const int* __restrict__ fg,
                         const float* __restrict__ sobj,
                         const float* __restrict__ negsum,
                         int* __restrict__ mcnt, int* __restrict__ ming,
                         int Bn, int Gn, int An, int NC) {
    __shared__ float scost[32 * KK];
    __shared__ int   sidx[32 * KK];
    __shared__ float siou[32 * KK];

    int blk = blockIdx.x;
    int b = blk / Gn, g = blk - b * Gn;
    int lane = threadIdx.x;

    const float* L = labels + ((size_t)b * Gn + g) * 5;
    int gcls = (int)L[0];
    float gx = L[1], gy = L[2], gw = L[3], gh = L[4];

    float kc_[KK]; int ki_[KK]; float kio[KK];
#pragma unroll
    for (int i = 0; i < KK; ++i) { kc_[i] = 3e38f; ki_[i] = 0x7fffffff; kio[i] = -1.0f; }

    for (int a = lane; a < An; a += 32) {
        // prefetch next iteration's strided cls-logit gather (global_prefetch_b8)
        int an = a + 32;
        if (an < An)
            __builtin_prefetch(&clsp[(size_t)(b * An + an) * NC + gcls], 0, 1);

        float s = strides[a];
        float cx = (xs[a] + 0.5f) * s, cy = (ys[a] + 0.5f) * s;
        float4 pb = ((const float4*)bbox)[(size_t)b * An + a];
        int t = b * An + a;
        int fga = fg[t];
        float ns = negsum[t];
        float so = sobj[t];
        float cl = clsp[(size_t)t * NC + gcls];
        float iou;
        float c = pair_cost(gx, gy, gw, gh, pb.x, pb.y, pb.z, pb.w,
                            cx, cy, s, ns, so, cl, fga, &iou);
        // min-cost top-K, (cost, idx) lexicographic ascending
        if (c < kc_[KK - 1] || (c == kc_[KK - 1] && a < ki_[KK - 1])) {
            kc_[KK - 1] = c; ki_[KK - 1] = a;
#pragma unroll
            for (int j = KK - 1; j > 0; --j) {
                bool sw = (kc_[j] < kc_[j - 1]) ||
                          (kc_[j] == kc_[j - 1] && ki_[j] < ki_[j - 1]);
                if (sw) {
                    float tc = kc_[j]; kc_[j] = kc_[j - 1]; kc_[j - 1] = tc;
                    int ti = ki_[j]; ki_[j] = ki_[j - 1]; ki_[j - 1] = ti;
                }
            }
        }
        // max-iou top-K of ious_fg
        float iv = fga ? iou : 0.0f;
        if (iv > kio[KK - 1]) {
            kio[KK - 1] = iv;
#pragma unroll
            for (int j = KK - 1; j > 0; --j) {
                if (kio[j] > kio[j - 1]) {
                    float tv = kio[j]; kio[j] = kio[j - 1]; kio[j - 1] = tv;
                }
            }
        }
    }

#pragma unroll
    for (int i = 0; i < KK; ++i) {
        scost[lane * KK + i] = kc_[i];
        sidx[lane * KK + i]  = ki_[i];
        siou[lane * KK + i]  = kio[i];
    }
    __syncthreads();

    if (lane == 0) {
        float fc[KK]; int fi[KK]; float fo[KK];
#pragma unroll
        for (int i = 0; i < KK; ++i) { fc[i] = 3e38f; fi[i] = 0x7fffffff; fo[i] = -1.0f; }
#pragma unroll 1
        for (int t = 0; t < 32; ++t) {
#pragma unroll 1
            for (int k = 0; k < KK; ++k) {
                float c = scost[t * KK + k];
                int ix = sidx[t * KK + k];
                if (ix != 0x7fffffff) {
                    if (c < fc[KK - 1] || (c == fc[KK - 1] && ix < fi[KK - 1])) {
                        fc[KK - 1] = c; fi[KK - 1] = ix;
#pragma unroll 1
                        for (int j = KK - 1; j > 0; --j) {
                            bool sw = (fc[j] < fc[j - 1]) ||
                                      (fc[j] == fc[j - 1] && fi[j] < fi[j - 1]);
                            if (sw) {
                                float tc = fc[j]; fc[j] = fc[j - 1]; fc[j - 1] = tc;
                                int ti = fi[j]; fi[j] = fi[j - 1]; fi[j - 1] = ti;
                            }
                        }
                    }
                }
                float v = siou[t * KK + k];
                if (v > fo[KK - 1]) {
                    fo[KK - 1] = v;
#pragma unroll 1
                    for (int j = KK - 1; j > 0; --j) {
                        if (fo[j] > fo[j - 1]) {
                            float tv = fo[j]; fo[j] = fo[j - 1]; fo[j - 1] = tv;
                        }
                    }
                }
            }
        }
        float ssum = 0.0f;
#pragma unroll
        for (int k = 0; k < KK; ++k) ssum += fmaxf(fo[k], 0.0f);
        int dk = (int)ssum;            // truncation, like .astype(int32)
        if (dk < 1) dk = 1;
        if (dk > KK) dk = KK;
#pragma unroll 1
        for (int k = 0; k < dk; ++k) {
            int a = fi[k];
            if (a >= 0 && a < An) {
                atomicAdd(&mcnt[b * An + a], 1);
                atomicMin(&ming[b * An + a], g);
            }
        }
    }
}

// ---------------------------------------------------------------------------
// Kernel D: resolve duplicates (first-argmin over cost column, identical
// formula), compute matched IoU, write all four outputs (fully overwrites
// d_out every call).
// ---------------------------------------------------------------------------
__global__ void k_output(const float* __restrict__ labels,
                         const float* __restrict__ bbox,
                         const float* __restrict__ clsp,
                         const float* __restrict__ strides,
                         const float* __restrict__ xs,
                         const float* __restrict__ ys,
                         const int* __restrict__ fg,
                         const float* __restrict__ sobj,
                         const float* __restrict__ negsum,
                         const int* __restrict__ mcnt,
                         const int* __restrict__ ming,
                         float* __restrict__ out,
                         int Bn, int Gn, int An, int NC) {
    int t = blockIdx.x * blockDim.x + threadIdx.x;
    if (t >= Bn * An) return;
    int b = t / An, a = t - b * An;

    float* cls_out = out;
    float* reg_out = out + (size_t)Bn * An * NC;
    float* obj_out = reg_out + (size_t)Bn * An * 4;
    float* fg_out  = obj_out + (size_t)Bn * An;
    size_t clsbase = (size_t)t * NC;

    int cnt = mcnt[t];
    if (cnt == 0) {
        for (int c = 0; c < NC; ++c) cls_out[clsbase + c] = 0.0f;
        reg_out[(size_t)t * 4 + 0] = 0.0f;
        reg_out[(size_t)t * 4 + 1] = 0.0f;
        reg_out[(size_t)t * 4 + 2] = 0.0f;
        reg_out[(size_t)t * 4 + 3] = 0.0f;
        obj_out[t] = 0.0f;
        fg_out[t] = 0.0f;
        return;
    }

    float4 pb = ((const float4*)bbox)[(size_t)b * An + a];
    int mg;
    if (cnt == 1) {
        mg = ming[t];
    } else {
        // duplicate anchor: only_min = first argmin over g of cost column
        float s = strides[a];
        float cx = (xs[a] + 0.5f) * s, cy = (ys[a] + 0.5f) * s;
        int fga = fg[t];
        float ns = negsum[t];
        float so = sobj[t];
        const float* L = labels + (size_t)b * Gn * 5;
        float best = 3e38f;
        mg = 0;
#pragma unroll 1
        for (int g = 0; g < Gn; ++g) {
            int gc = (int)L[g * 5 + 0];
            float cl = clsp[clsbase + gc];
            float iou;
            float c = pair_cost(L[g * 5 + 1], L[g * 5 + 2], L[g * 5 + 3], L[g * 5 + 4],
                                pb.x, pb.y, pb.z, pb.w, cx, cy, s,
                                ns, so, cl, fga, &iou);
            if (c < best) { best = c; mg = g; }
        }
    }

    const float* Lm = labels + ((size_t)b * Gn + mg) * 5;
    int gcls = (int)Lm[0];
    float piou = pair_iou(Lm[1], Lm[2], Lm[3], Lm[4], pb.x, pb.y, pb.z, pb.w);

    for (int c = 0; c < NC; ++c) cls_out[clsbase + c] = 0.0f;
    if (gcls >= 0 && gcls < NC) cls_out[clsbase + gcls] = piou;
    reg_out[(size_t)t * 4 + 0] = Lm[1];
    reg_out[(size_t)t * 4 + 1] = Lm[2];
    reg_out[(size_t)t * 4 + 2] = Lm[3];
    reg_out[(size_t)t * 4 + 3] = Lm[4];
    obj_out[t] = 1.0f;
    fg_out[t] = 1.0f;
}

extern "C" void kernel_launch(void* const* d_in, const int* in_sizes, int n_in,
                              void* d_out, int out_size, void* d_ws, size_t ws_size,
                              hipStream_t stream) {
    const float* labels  = (const float*)d_in[0];
    const float* bbox    = (const float*)d_in[1];
    const float* clsp    = (const float*)d_in[2];
    const float* objp    = (const float*)d_in[3];
    const float* strides = (const float*)d_in[4];
    const float* xs      = (const float*)d_in[5];
    const float* ys      = (const float*)d_in[6];

    int An = in_sizes[4];                 // expanded_strides is (1, A)
    int Bn = in_sizes[1] / (An * 4);      // bbox_preds (B, A, 4)
    int Gn = in_sizes[0] / (Bn * 5);      // labels (B, G, 5)
    int NC = in_sizes[2] / (Bn * An);     // cls_preds (B, A, NC)
    int BA = Bn * An;

    char* ws = (char*)d_ws;
    int*   fgw    = (int*)ws;
    float* sobj   = (float*)(ws + (size_t)BA * 4);
    float* negsum = (float*)(ws + (size_t)BA * 8);
    int*   mcnt   = (int*)(ws + (size_t)BA * 12);
    int*   ming   = (int*)(ws + (size_t)BA * 16);
    float* out = (float*)d_out;

    k_precompute<<<(BA + 255) / 256, 256, 0, stream>>>(
        labels, objp, strides, xs, ys, fgw, sobj, mcnt, ming, Bn, Gn, An);

    int tiles = (An + 15) / 16;
    k_negsum<<<Bn * tiles, 32, 0, stream>>>(clsp, sobj, negsum, Bn, An, NC);

    k_assign<<<Bn * Gn, 32, 0, stream>>>(
        labels, bbox, clsp, strides, xs, ys, fgw, sobj, negsum, mcnt, ming,
        Bn, Gn, An, NC);

    k_output<<<(BA + 255) / 256, 256, 0, stream>>>(
        labels, bbox, clsp, strides, xs, ys, fgw, sobj, negsum, mcnt, ming, out,
        Bn, Gn, An, NC);
}